// VectorAggregate_81733227643030
// MI455X (gfx1250) — compile-verified
//
#include <hip/hip_runtime.h>
#include <math.h>

#ifndef __has_builtin
#define __has_builtin(x) 0
#endif

#if __has_builtin(__builtin_amdgcn_global_load_async_to_lds_b32)
#define USE_ASYNC_BUILTIN 1
#else
#define USE_ASYNC_BUILTIN 0
#endif

#if __has_builtin(__builtin_amdgcn_s_wait_asynccnt)
#define USE_WAIT_BUILTIN 1
#else
#define USE_WAIT_BUILTIN 0
#endif

static constexpr int kC  = 32;
static constexpr int kH  = 128;
static constexpr int kW  = 160;
static constexpr int kD  = 48;
static constexpr int kS  = 2;
static constexpr int kG  = 8;
static constexpr int kHW = kH * kW;
static constexpr int kBlock  = 256;
static constexpr int kDChunk = 4;   // more depth chunks -> more waves in flight

typedef __attribute__((address_space(1))) int  gint;   // global addrspace int
typedef __attribute__((address_space(3))) int  lint;   // LDS addrspace int
typedef __attribute__((address_space(3))) float lds_float;

__device__ __forceinline__ float sigmoidf_(float x) {
  return 1.0f / (1.0f + __expf(-x));
}

// ---------------------------------------------------------------------------
// Setup: rt[s] = rows 0..2 of (src_proj[s] @ inv(ref_proj)) -> 9 rot + 3 trans
// ---------------------------------------------------------------------------
__global__ void setup_proj_kernel(const float* __restrict__ ref_proj,
                                  const float* __restrict__ src_projs,
                                  float* __restrict__ rt)
{
  if (threadIdx.x != 0 || blockIdx.x != 0) return;
  // Gauss-Jordan 4x4 inverse with partial pivoting
  float a[4][8];
  for (int i = 0; i < 4; ++i)
    for (int j = 0; j < 4; ++j) {
      a[i][j]     = ref_proj[i * 4 + j];
      a[i][4 + j] = (i == j) ? 1.0f : 0.0f;
    }
  for (int col = 0; col < 4; ++col) {
    int piv = col; float best = fabsf(a[col][col]);
    for (int r = col + 1; r < 4; ++r) {
      float v = fabsf(a[r][col]);
      if (v > best) { best = v; piv = r; }
    }
    if (piv != col)
      for (int j = 0; j < 8; ++j) { float tv = a[col][j]; a[col][j] = a[piv][j]; a[piv][j] = tv; }
    float ip = 1.0f / a[col][col];
    for (int j = 0; j < 8; ++j) a[col][j] *= ip;
    for (int r = 0; r < 4; ++r) {
      if (r == col) continue;
      float f = a[r][col];
      for (int j = 0; j < 8; ++j) a[r][j] -= f * a[col][j];
    }
  }
  for (int s = 0; s < kS; ++s) {
    const float* P = src_projs + s * 16;
    for (int i = 0; i < 3; ++i) {
      float m[4];
      for (int j = 0; j < 4; ++j) {
        float acc = 0.0f;
        for (int k = 0; k < 4; ++k) acc += P[i * 4 + k] * a[k][4 + j];
        m[j] = acc;
      }
      rt[s * 12 + i * 3 + 0] = m[0];
      rt[s * 12 + i * 3 + 1] = m[1];
      rt[s * 12 + i * 3 + 2] = m[2];
      rt[s * 12 + 9 + i]     = m[3];
    }
  }
}

// ---------------------------------------------------------------------------
// Fused: warp + group softmax correlation + depth-weight MLP + blend
// grid.x = HW/256 (pixel tiles), grid.y = D/kDChunk (depth chunks)
// ---------------------------------------------------------------------------
__global__ __launch_bounds__(kBlock)
void fused_cost_volume_kernel(const float* __restrict__ ref,    // (C,H,W)
                              const float* __restrict__ srcs,   // (S,C,H,W)
                              const float* __restrict__ depth,  // (D)
                              const float* __restrict__ rt,     // (S,12)
                              const float* __restrict__ w1,     // (G)
                              const float* __restrict__ bn_gamma,
                              const float* __restrict__ bn_beta,
                              const float* __restrict__ bn_mean,
                              const float* __restrict__ bn_var,
                              const float* __restrict__ w2,
                              const float* __restrict__ b2,
                              float* __restrict__ out)          // (G,D,H,W)
{
  __shared__ float ldsRef[kC * kBlock];

  const int t = threadIdx.x;
  const int p = blockIdx.x * kBlock + t;
  if (p >= kHW) return;

  // Pre-warm both source feature maps into cache (global_prefetch_b8)
  __builtin_prefetch(srcs + p, 0, 1);
  __builtin_prefetch(srcs + kC * kHW + p, 0, 1);

  // ---- Stage this tile's reference features into LDS (async DMA path) ----
  #pragma unroll
  for (int c = 0; c < kC; ++c) {
#if USE_ASYNC_BUILTIN
    __builtin_amdgcn_global_load_async_to_lds_b32(
        (gint*)(ref + (size_t)c * kHW + p),
        (lint*)&ldsRef[c * kBlock + t], 0, 0);
#else
    unsigned long long ga = (unsigned long long)(const void*)(ref + (size_t)c * kHW + p);
    unsigned la = (unsigned)(unsigned long long)(lds_float*)&ldsRef[c * kBlock + t];
    asm volatile("global_load_async_to_lds_b32 %0, %1, off"
                 :: "v"(la), "v"(ga) : "memory");
#endif
  }
#if USE_WAIT_BUILTIN
  __builtin_amdgcn_s_wait_asynccnt(0);
#else
  asm volatile("s_wait_asynccnt 0" ::: "memory");
#endif
  __syncthreads();

  // ---- Reference group softmax (computed once, reused kDChunk*S times) ----
  float refp[kC];
  #pragma unroll
  for (int g = 0; g < kG; ++g) {
    float r0 = ldsRef[(g * 4 + 0) * kBlock + t];
    float r1 = ldsRef[(g * 4 + 1) * kBlock + t];
    float r2 = ldsRef[(g * 4 + 2) * kBlock + t];
    float r3 = ldsRef[(g * 4 + 3) * kBlock + t];
    float m  = fmaxf(fmaxf(r0, r1), fmaxf(r2, r3));
    float e0 = __expf(r0 - m), e1 = __expf(r1 - m);
    float e2 = __expf(r2 - m), e3 = __expf(r3 - m);
    float is = 1.0f / (e0 + e1 + e2 + e3);
    refp[g * 4 + 0] = e0 * is;
    refp[g * 4 + 1] = e1 * is;
    refp[g * 4 + 2] = e2 * is;
    refp[g * 4 + 3] = e3 * is;
  }

  // ---- Uniform parameters into registers ----
  float rtv[kS][12];
  #pragma unroll
  for (int s = 0; s < kS; ++s)
    #pragma unroll
    for (int i = 0; i < 12; ++i) rtv[s][i] = rt[s * 12 + i];

  float w1r[kG];
  #pragma unroll
  for (int g = 0; g < kG; ++g) w1r[g] = w1[g];
  const float bscale = bn_gamma[0] / sqrtf(bn_var[0] + 1e-5f);
  const float bbias  = bn_beta[0] - bn_mean[0] * bscale;
  const float w2v = w2[0], b2v = b2[0];

  const float fx = (float)(p % kW);
  const float fy = (float)(p / kW);

  for (int dd = 0; dd < kDChunk; ++dd) {
    const int d = blockIdx.y * kDChunk + dd;
    const float dep = depth[d];

    float vol[kG];
    #pragma unroll
    for (int g = 0; g < kG; ++g) vol[g] = 0.0f;
    float wsum = 0.0f;

    #pragma unroll
    for (int s = 0; s < kS; ++s) {
      const float* rts = rtv[s];
      const float* sf  = srcs + (size_t)s * kC * kHW;

      float pxn = (rts[0] * fx + rts[1] * fy + rts[2]) * dep + rts[9];
      float pyn = (rts[3] * fx + rts[4] * fy + rts[5]) * dep + rts[10];
      float pzn = (rts[6] * fx + rts[7] * fy + rts[8]) * dep + rts[11];
      float iz  = 1.0f / pzn;
      float px  = pxn * iz, py = pyn * iz;

      float x0f = floorf(px), y0f = floorf(py);
      float wxf = px - x0f,  wyf = py - y0f;
      float x1f = x0f + 1.0f, y1f = y0f + 1.0f;

      bool vx0 = (x0f >= 0.0f) && (x0f <= (float)(kW - 1));
      bool vx1 = (x1f >= 0.0f) && (x1f <= (float)(kW - 1));
      bool vy0 = (y0f >= 0.0f) && (y0f <= (float)(kH - 1));
      bool vy1 = (y1f >= 0.0f) && (y1f <= (float)(kH - 1));

      int xi0 = (int)fminf(fmaxf(x0f, 0.0f), (float)(kW - 1));
      int xi1 = (int)fminf(fmaxf(x1f, 0.0f), (float)(kW - 1));
      int yi0 = (int)fminf(fmaxf(y0f, 0.0f), (float)(kH - 1));
      int yi1 = (int)fminf(fmaxf(y1f, 0.0f), (float)(kH - 1));

      float w00 = (1.0f - wxf) * (1.0f - wyf) * ((vx0 && vy0) ? 1.0f : 0.0f);
      float w10 = wxf          * (1.0f - wyf) * ((vx1 && vy0) ? 1.0f : 0.0f);
      float w01 = (1.0f - wxf) * wyf          * ((vx0 && vy1) ? 1.0f : 0.0f);
      float w11 = wxf          * wyf          * ((vx1 && vy1) ? 1.0f : 0.0f);

      int o00 = yi0 * kW + xi0, o10 = yi0 * kW + xi1;
      int o01 = yi1 * kW + xi0, o11 = yi1 * kW + xi1;

      float vg[kG];
      float lin = 0.0f;
      #pragma unroll
      for (int g = 0; g < kG; ++g) {
        float tc0, tc1, tc2, tc3;
        {
          const float* fp = sf + (size_t)(g * 4 + 0) * kHW;
          tc0 = w00 * fp[o00] + w10 * fp[o10] + w01 * fp[o01] + w11 * fp[o11];
        }
        {
          const float* fp = sf + (size_t)(g * 4 + 1) * kHW;
          tc1 = w00 * fp[o00] + w10 * fp[o10] + w01 * fp[o01] + w11 * fp[o11];
        }
        {
          const float* fp = sf + (size_t)(g * 4 + 2) * kHW;
          tc2 = w00 * fp[o00] + w10 * fp[o10] + w01 * fp[o01] + w11 * fp[o11];
        }
        {
          const float* fp = sf + (size_t)(g * 4 + 3) * kHW;
          tc3 = w00 * fp[o00] + w10 * fp[o10] + w01 * fp[o01] + w11 * fp[o11];
        }
        float m  = fmaxf(fmaxf(tc0, tc1), fmaxf(tc2, tc3));
        float e0 = __expf(tc0 - m), e1 = __expf(tc1 - m);
        float e2 = __expf(tc2 - m), e3 = __expf(tc3 - m);
        float is = 1.0f / (e0 + e1 + e2 + e3);
        float v  = (e0 * refp[g * 4 + 0] + e1 * refp[g * 4 + 1] +
                    e2 * refp[g * 4 + 2] + e3 * refp[g * 4 + 3]) * is;
        vg[g] = v;
        lin  += w1r[g] * v;
      }

      float xw = lin * bscale + bbias;   // BatchNorm (inference form)
      xw = fmaxf(xw, 0.0f);              // ReLU
      xw = xw * w2v + b2v;               // 1x1 conv
      float wgt = sigmoidf_(xw);         // sigmoid

      wsum += wgt;
      #pragma unroll
      for (int g = 0; g < kG; ++g) vol[g] += wgt * vg[g];
    }

    float iw = 1.0f / wsum;
    #pragma unroll
    for (int g = 0; g < kG; ++g)
      out[(size_t)g * (kD * kHW) + (size_t)d * kHW + p] = vol[g] * iw;
  }
}

// ---------------------------------------------------------------------------
extern "C" void kernel_launch(void* const* d_in, const int* in_sizes, int n_in,
                              void* d_out, int out_size, void* d_ws, size_t ws_size,
                              hipStream_t stream)
{
  (void)in_sizes; (void)n_in; (void)out_size; (void)ws_size;

  const float* ref   = (const float*)d_in[0];   // (1,32,128,160)
  const float* srcs  = (const float*)d_in[1];   // (2,1,32,128,160)
  const float* refP  = (const float*)d_in[2];   // (1,4,4)
  const float* srcP  = (const float*)d_in[3];   // (2,1,4,4)
  const float* depth = (const float*)d_in[4];   // (1,48)
  const float* w1    = (const float*)d_in[5];   // (1,8)
  const float* bg    = (const float*)d_in[6];
  const float* bb    = (const float*)d_in[7];
  const float* bm    = (const float*)d_in[8];
  const float* bv    = (const float*)d_in[9];
  const float* w2    = (const float*)d_in[10];
  const float* b2    = (const float*)d_in[11];
  float* out = (float*)d_out;                   // (1,8,48,128,160)
  float* rt  = (float*)d_ws;                    // 24 floats

  setup_proj_kernel<<<dim3(1), dim3(32), 0, stream>>>(refP, srcP, rt);

  dim3 grid(kHW / kBlock, kD / kDChunk);        // 80 x 12
  fused_cost_volume_kernel<<<grid, dim3(kBlock), 0, stream>>>(
      ref, srcs, depth, rt, w1, bg, bb, bm, bv, w2, b2, out);
}